// DataLoss_9878424781365
// MI455X (gfx1250) — compile-verified
//
#include <hip/hip_runtime.h>

typedef __attribute__((ext_vector_type(2))) float v2f;
typedef __attribute__((ext_vector_type(4))) float v4f;
typedef __attribute__((ext_vector_type(8))) float v8f;
typedef __attribute__((ext_vector_type(4))) int   v4i;

#define CL_SCALE  64.0f
#define CL_MARGIN 0.5f

// loss_i = logaddexp(lp, ln) - lp = softplus(ln - lp)
//        = max(d,0) + log(1 + exp(-|d|)),  d = ln - lp
// Native transcendentals: args to cos are bounded (|x| <~ 6 rad), so
// v_cos_f32 phase error ~4e-7 rad -> negligible vs the f32 sum tolerance.
__device__ __forceinline__ float sample_loss(float a, float b, int y) {
    // label==0 -> pos=a, neg=b ; label==1 -> pos=b, neg=a
    float pos = y ? b : a;
    float neg = y ? a : b;
    float lp = CL_SCALE * __cosf(pos + CL_MARGIN);
    float ln = CL_SCALE * __cosf(neg);
    float d  = ln - lp;
    float t  = __expf(-fabsf(d));
    return fmaxf(d, 0.0f) + __logf(1.0f + t);
}

// Stage 1: streaming compute + per-block partial (double) into d_ws.
// Wave-level reduction done with V_WMMA_F32_16X16X4_F32 (B = ones).
__global__ __launch_bounds__(256) void cosface_partial_kernel(
    const float* __restrict__ dist,   // (2, n) row-major
    const int*   __restrict__ label,  // (n,)
    double*      __restrict__ partials,
    int n) {
    const int tid      = blockIdx.x * blockDim.x + threadIdx.x;
    const int nthreads = gridDim.x * blockDim.x;
    const int nv4      = n >> 2;

    const v4f* __restrict__ r0 = (const v4f*)dist;
    const v4f* __restrict__ r1 = (const v4f*)(dist + n);
    const v4i* __restrict__ lb = (const v4i*)label;

    float acc = 0.0f;
    for (int i = tid; i < nv4; i += nthreads) {
        // single-pass streaming data (~200MB ~ L2 size): non-temporal loads
        v4f a = __builtin_nontemporal_load(&r0[i]);
        v4f b = __builtin_nontemporal_load(&r1[i]);
        v4i y = __builtin_nontemporal_load(&lb[i]);
        acc += sample_loss(a.x, b.x, y.x);
        acc += sample_loss(a.y, b.y, y.y);
        acc += sample_loss(a.z, b.z, y.z);
        acc += sample_loss(a.w, b.w, y.w);
    }
    // tail (n not multiple of 4) — handled by first few threads, then reconverge
    for (int i = (nv4 << 2) + tid; i < n; i += nthreads) {
        acc += sample_loss(dist[i], dist[n + i], label[i]);
    }

    // ---- wave32 reduction via WMMA: D = A(16x4) * ones(4x16) + 0 ----
    // A layout (ISA 7.12.2): lanes 0-15 -> A(m,0)=VGPR0, A(m,1)=VGPR1 (m=lane)
    //                        lanes 16-31 -> A(m,2)=VGPR0, A(m,3)=VGPR1 (m=lane-16)
    // With A.y = 0 and B = all-ones (layout-independent):
    //   D(m, n) = acc[lane m] + acc[lane m+16]  for every n.
    v2f A; A.x = acc;  A.y = 0.0f;
    v2f B; B.x = 1.0f; B.y = 1.0f;
    v8f C = {0.0f, 0.0f, 0.0f, 0.0f, 0.0f, 0.0f, 0.0f, 0.0f};
    v8f D = __builtin_amdgcn_wmma_f32_16x16x4_f32(
        /*neg_a=*/false, A, /*neg_b=*/false, B,
        /*c_mod=*/(short)0, C, /*reuse_a=*/false, /*reuse_b=*/false);

    // C/D layout: VGPR r holds row r (lanes 0-15) or row 8+r (lanes 16-31).
    // Summing the 8 regs: lanes<16 hold sum of rows 0-7, lanes>=16 rows 8-15.
    float part = D[0] + D[1] + D[2] + D[3] + D[4] + D[5] + D[6] + D[7];
    float wave_sum = part + __shfl_xor(part, 16, 32);

    __shared__ float wsum[8];
    const int lane = threadIdx.x & 31;
    const int wid  = threadIdx.x >> 5;
    if (lane == 0) wsum[wid] = wave_sum;
    __syncthreads();

    if (threadIdx.x == 0) {
        double s = 0.0;
        const int nw = blockDim.x >> 5;
        for (int w = 0; w < nw; ++w) s += (double)wsum[w];
        partials[blockIdx.x] = s;
    }
}

// Stage 2: deterministic single-block double reduction -> f32 scalar.
__global__ __launch_bounds__(256) void reduce_partials_kernel(
    const double* __restrict__ partials, int nparts, float* __restrict__ out) {
    __shared__ double s[256];
    double a = 0.0;
    for (int i = threadIdx.x; i < nparts; i += 256) a += partials[i];
    s[threadIdx.x] = a;
    __syncthreads();
    for (int ofs = 128; ofs > 0; ofs >>= 1) {
        if (threadIdx.x < ofs) s[threadIdx.x] += s[threadIdx.x + ofs];
        __syncthreads();
    }
    if (threadIdx.x == 0) out[0] = (float)s[0];
}

extern "C" void kernel_launch(void* const* d_in, const int* in_sizes, int n_in,
                              void* d_out, int out_size, void* d_ws, size_t ws_size,
                              hipStream_t stream) {
    const float* dist  = (const float*)d_in[0];  // (2, N) float32
    const int*   label = (const int*)d_in[1];    // (N,) int32
    const int    n     = in_sizes[1];
    float*  out      = (float*)d_out;
    double* partials = (double*)d_ws;            // 2048 * 8B = 16 KB scratch

    const int threads = 256;
    const int blocks  = 2048;                    // persistent grid-stride, 8 vec4 iters/thread

    cosface_partial_kernel<<<blocks, threads, 0, stream>>>(dist, label, partials, n);
    reduce_partials_kernel<<<1, threads, 0, stream>>>(partials, blocks, out);
}